// Edgeconv_fw_43679817400588
// MI455X (gfx1250) — compile-verified
//
#include <hip/hip_runtime.h>
#include <math.h>

// ---------------------------------------------------------------------------
// EdgeConv forward for MI455X (gfx1250, wave32).
//   B=16, C=3, N=2048, K=20 neighbors, OUT=64 channels, IN=2*C=6 channels.
// Strategy: never materialize y (B,64,N,20) = 671MB of f32 traffic. Store the
// kNN idx (2.6MB) in workspace and run the 6->64 GEMM twice with
// V_WMMA_F32_16X16X4_F32 (K padded 6->8, two WMMAs per 16x16 tile):
//   pass 1 accumulates BN batch statistics (sum, sumsq per channel),
//   pass 2 recomputes y fused with affine + ReLU + max-over-k.
// Total HBM traffic ~15MB -> sub-microsecond at 23.3 TB/s; kNN selection is
// the real bottleneck, so its LDS inner loop is vectorized to b128 broadcasts.
// ---------------------------------------------------------------------------

#define BB 16
#define NN 2048
#define KK 20
#define OUTC 64
#define TOTP (BB * NN * KK)      // 655360 conv points
#define NGROUP (TOTP / 16)       // 40960 groups of 16 points

typedef __attribute__((ext_vector_type(2))) float v2f;
typedef __attribute__((ext_vector_type(8))) float v8f;

// D = A(16x4,f32) * B(4x16,f32) + C(16x16,f32)
__device__ __forceinline__ v8f wmma_f32_16x16x4(v2f a, v2f b, v8f c) {
  return __builtin_amdgcn_wmma_f32_16x16x4_f32(
      /*neg_a=*/false, a, /*neg_b=*/false, b,
      /*c_mod=*/(short)0, c, /*reuse_a=*/false, /*reuse_b=*/false);
}

// A-tile layout (16x4 f32): lane = M (mod 16); VGPR0 holds K=0 (lanes 0-15) /
// K=2 (lanes 16-31); VGPR1 holds K=1 / K=3.  W is (64 x 6) row-major.
__device__ __forceinline__ void load_a_tiles(const float* __restrict__ Wm,
                                             int lm, int halfHi, v2f a[4][2]) {
#pragma unroll
  for (int t = 0; t < 4; ++t) {
#pragma unroll
    for (int s = 0; s < 2; ++s) {
      const int row = t * 16 + lm;
      const int k0 = s * 4 + (halfHi ? 2 : 0);
      const int k1 = s * 4 + (halfHi ? 3 : 1);
      a[t][s].x = (k0 < 6) ? Wm[row * 6 + k0] : 0.0f;
      a[t][s].y = (k1 < 6) ? Wm[row * 6 + k1] : 0.0f;
    }
  }
}

// B-tile (4x16 f32) for one group of 16 points: lane column = lane%16.
// VGPR0 = K rows {0,2}, VGPR1 = K rows {1,3} (lo/hi half).  feat channels:
// [f0,f1,f2, xi0,xi1,xi2, 0, 0] with f = x[j]-x[i] (neighbor - center).
__device__ __forceinline__ void make_b_tiles(float f0, float f1, float f2,
                                             float xi0, float xi1, float xi2,
                                             int halfHi, v2f& b0, v2f& b1) {
  b0.x = halfHi ? f2 : f0;    // K=2 (ch2) : K=0 (ch0)
  b0.y = halfHi ? xi0 : f1;   // K=3 (ch3) : K=1 (ch1)
  b1.x = halfHi ? 0.f : xi1;  // K=6 (pad) : K=4 (ch4)
  b1.y = halfHi ? 0.f : xi2;  // K=7 (pad) : K=5 (ch5)
}

// Stable top-K insert (sorted descending, register resident, fully unrolled).
// Strict '>' keeps earlier indices on ties -> matches jax.lax.top_k order.
__device__ __forceinline__ void topk_push(float d, int j,
                                          float (&topd)[KK], int (&topi)[KK],
                                          float& kth) {
  if (d > kth) {
    float dv = d;
    int ji = j;
#pragma unroll
    for (int m = 0; m < KK; ++m) {
      const bool gt = dv > topd[m];
      const float td = topd[m];
      const int ti = topi[m];
      topd[m] = gt ? dv : td;
      topi[m] = gt ? ji : ti;
      dv = gt ? td : dv;
      ji = gt ? ti : ji;
    }
    kth = topd[KK - 1];
  }
}

// ---------------------------------------------------------------------------
// Kernel 1: kNN.  One block = 256 rows of one batch; x staged in LDS (24KB of
// the 320KB WGP pool).  Inner loop reads candidates 4-wide per channel via
// ds_load_b128 broadcasts (all lanes share the address), cutting DS issue 4x.
// Self-distance 0 is the maximum of -||xi-xj||^2, as in the reference.
// ---------------------------------------------------------------------------
__global__ void knn_kernel(const float* __restrict__ x, int* __restrict__ idxbuf) {
  __shared__ float xs[3 * NN];
  const int b = blockIdx.x >> 3;   // 8 row-blocks per batch
  const int rb = blockIdx.x & 7;
  const float* xb = x + (size_t)b * 3 * NN;
  for (int t = threadIdx.x; t < 3 * NN; t += blockDim.x) xs[t] = xb[t];
  __syncthreads();

  const int row = rb * 256 + threadIdx.x;
  const float xi0 = xs[row], xi1 = xs[NN + row], xi2 = xs[2 * NN + row];

  float topd[KK];
  int topi[KK];
#pragma unroll
  for (int m = 0; m < KK; ++m) { topd[m] = -INFINITY; topi[m] = 0; }
  float kth = -INFINITY;

  const float4* c0 = (const float4*)(xs);
  const float4* c1 = (const float4*)(xs + NN);
  const float4* c2 = (const float4*)(xs + 2 * NN);

  for (int j4 = 0; j4 < NN / 4; ++j4) {
    const float4 a0 = c0[j4];
    const float4 a1 = c1[j4];
    const float4 a2 = c2[j4];
    const int jb = j4 * 4;

    float d0, d1, d2, d;
    d0 = a0.x - xi0; d1 = a1.x - xi1; d2 = a2.x - xi2;
    d = -(d0 * d0 + d1 * d1 + d2 * d2);
    topk_push(d, jb + 0, topd, topi, kth);

    d0 = a0.y - xi0; d1 = a1.y - xi1; d2 = a2.y - xi2;
    d = -(d0 * d0 + d1 * d1 + d2 * d2);
    topk_push(d, jb + 1, topd, topi, kth);

    d0 = a0.z - xi0; d1 = a1.z - xi1; d2 = a2.z - xi2;
    d = -(d0 * d0 + d1 * d1 + d2 * d2);
    topk_push(d, jb + 2, topd, topi, kth);

    d0 = a0.w - xi0; d1 = a1.w - xi1; d2 = a2.w - xi2;
    d = -(d0 * d0 + d1 * d1 + d2 * d2);
    topk_push(d, jb + 3, topd, topi, kth);
  }

  int* op = idxbuf + ((size_t)b * NN + row) * KK;
#pragma unroll
  for (int m = 0; m < KK; ++m) op[m] = topi[m];
}

// ---------------------------------------------------------------------------
// Kernel 2: zero the statistics accumulators.
// ---------------------------------------------------------------------------
__global__ void zero_kernel(float* __restrict__ p, int n) {
  const int i = blockIdx.x * blockDim.x + threadIdx.x;
  if (i < n) p[i] = 0.0f;
}

// ---------------------------------------------------------------------------
// Kernel 3: BN statistics.  Each wave sweeps groups of 16 conv points,
// computes y(64x16) with 8 WMMAs, accumulates sum / sumsq per channel, then
// shuffle-reduces within each 16-lane half and atomically merges.
// ---------------------------------------------------------------------------
__global__ void stats_kernel(const float* __restrict__ x,
                             const float* __restrict__ Wm,
                             const int* __restrict__ idxbuf,
                             float* __restrict__ sums,
                             float* __restrict__ sumsq) {
  const int lane = threadIdx.x & 31;
  const int lm = lane & 15;
  const int halfHi = lane >> 4;
  const int waveId = (blockIdx.x * blockDim.x + threadIdx.x) >> 5;
  const int nWaves = (gridDim.x * blockDim.x) >> 5;

  v2f a[4][2];
  load_a_tiles(Wm, lm, halfHi, a);

  v8f sAcc[4], qAcc[4];
#pragma unroll
  for (int t = 0; t < 4; ++t) { sAcc[t] = (v8f){}; qAcc[t] = (v8f){}; }

  for (int g = waveId; g < NGROUP; g += nWaves) {
    const int p = g * 16 + lm;
    const int b = p / (NN * KK);
    const int rem = p - b * (NN * KK);
    const int n = rem / KK;
    const int j = idxbuf[p];
    const float* xb = x + (size_t)b * 3 * NN;
    const float xi0 = xb[n], xi1 = xb[NN + n], xi2 = xb[2 * NN + n];
    const float f0 = xb[j] - xi0, f1 = xb[NN + j] - xi1, f2 = xb[2 * NN + j] - xi2;

    v2f b0, b1;
    make_b_tiles(f0, f1, f2, xi0, xi1, xi2, halfHi, b0, b1);

#pragma unroll
    for (int t = 0; t < 4; ++t) {
      v8f d = wmma_f32_16x16x4(a[t][0], b0, (v8f){});
      d = wmma_f32_16x16x4(a[t][1], b1, d);
#pragma unroll
      for (int r = 0; r < 8; ++r) {
        sAcc[t][r] += d[r];
        qAcc[t][r] += d[r] * d[r];
      }
    }
  }

  // Reduce over the 16 N-columns of each half-wave; element (t,r,half) is
  // channel 16t + r + 8*half.
#pragma unroll
  for (int t = 0; t < 4; ++t) {
#pragma unroll
    for (int r = 0; r < 8; ++r) {
      float s = sAcc[t][r], q = qAcc[t][r];
#pragma unroll
      for (int m = 8; m >= 1; m >>= 1) {
        s += __shfl_xor(s, m, 32);
        q += __shfl_xor(q, m, 32);
      }
      if (lm == 0) {
        const int ch = t * 16 + r + (halfHi ? 8 : 0);
        atomicAdd(&sums[ch], s);
        atomicAdd(&sumsq[ch], q);
      }
    }
  }
}

// ---------------------------------------------------------------------------
// Kernel 4: fold BN into per-channel scale/shift.
// ---------------------------------------------------------------------------
__global__ void affine_kernel(const float* __restrict__ sums,
                              const float* __restrict__ sumsq,
                              const float* __restrict__ bnw,
                              const float* __restrict__ bnb,
                              float* __restrict__ scale,
                              float* __restrict__ shift) {
  const int c = threadIdx.x;
  if (c < OUTC) {
    const float inv = 1.0f / (float)TOTP;
    const float mean = sums[c] * inv;
    const float var = sumsq[c] * inv - mean * mean;  // biased, as reference
    const float sc = bnw[c] * rsqrtf(var + 1e-5f);
    scale[c] = sc;
    shift[c] = bnb[c] - mean * sc;
  }
}

// ---------------------------------------------------------------------------
// Kernel 5: output.  One wave owns 16 (b,n) points; for k=0..19 it rebuilds
// the feature tile, runs 8 WMMAs, applies affine+ReLU and keeps a running max
// directly in accumulator layout, then stores coalesced (16 lanes = 16
// consecutive n per channel row).
// ---------------------------------------------------------------------------
__global__ void out_kernel(const float* __restrict__ x,
                           const float* __restrict__ Wm,
                           const int* __restrict__ idxbuf,
                           const float* __restrict__ scale,
                           const float* __restrict__ shift,
                           float* __restrict__ out) {
  const int lane = threadIdx.x & 31;
  const int lm = lane & 15;
  const int halfHi = lane >> 4;
  const int waveId = (blockIdx.x * blockDim.x + threadIdx.x) >> 5;
  const int bnBase = waveId * 16;
  if (bnBase >= BB * NN) return;

  const int q = bnBase + lm;  // flat (b,n); 16-groups never straddle batches
  const int b = q / NN;
  const int n = q - b * NN;

  v2f a[4][2];
  load_a_tiles(Wm, lm, halfHi, a);

  float sc[4][8], sh[4][8];
#pragma unroll
  for (int t = 0; t < 4; ++t)
#pragma unroll
    for (int r = 0; r < 8; ++r) {
      const int ch = t * 16 + r + (halfHi ? 8 : 0);
      sc[t][r] = scale[ch];
      sh[t][r] = shift[ch];
    }

  float mx[4][8];
#pragma unroll
  for (int t = 0; t < 4; ++t)
#pragma unroll
    for (int r = 0; r < 8; ++r) mx[t][r] = 0.0f;  // max over k of relu(.) >= 0

  const float* xb = x + (size_t)b * 3 * NN;
  const float xi0 = xb[n], xi1 = xb[NN + n], xi2 = xb[2 * NN + n];
  const int* ip = idxbuf + (size_t)q * KK;

  for (int k = 0; k < KK; ++k) {
    const int j = ip[k];
    const float f0 = xb[j] - xi0, f1 = xb[NN + j] - xi1, f2 = xb[2 * NN + j] - xi2;
    v2f b0, b1;
    make_b_tiles(f0, f1, f2, xi0, xi1, xi2, halfHi, b0, b1);

#pragma unroll
    for (int t = 0; t < 4; ++t) {
      v8f d = wmma_f32_16x16x4(a[t][0], b0, (v8f){});
      d = wmma_f32_16x16x4(a[t][1], b1, d);
#pragma unroll
      for (int r = 0; r < 8; ++r) {
        float v = d[r] * sc[t][r] + sh[t][r];
        v = v > 0.0f ? v : 0.0f;
        mx[t][r] = fmaxf(mx[t][r], v);
      }
    }
  }

  // out is (B, 64, N): lane element (t,r) -> channel 16t+r+8*halfHi, point n.
#pragma unroll
  for (int t = 0; t < 4; ++t)
#pragma unroll
    for (int r = 0; r < 8; ++r) {
      const int ch = t * 16 + r + (halfHi ? 8 : 0);
      out[((size_t)b * OUTC + ch) * NN + n] = mx[t][r];
    }
}

// ---------------------------------------------------------------------------
extern "C" void kernel_launch(void* const* d_in, const int* in_sizes, int n_in,
                              void* d_out, int out_size, void* d_ws, size_t ws_size,
                              hipStream_t stream) {
  const float* x = (const float*)d_in[0];     // (16, 3, 2048)
  const float* Wm = (const float*)d_in[1];    // (64, 6)
  const float* bnw = (const float*)d_in[2];   // (64,)
  const float* bnb = (const float*)d_in[3];   // (64,)
  float* out = (float*)d_out;                 // (16, 64, 2048)

  int* idxbuf = (int*)d_ws;                            // 655360 ints
  float* stats = (float*)((char*)d_ws + (size_t)TOTP * sizeof(int));
  float* sums = stats;                                 // 64
  float* sumsq = stats + 64;                           // 64
  float* scale = stats + 128;                          // 64
  float* shift = stats + 192;                          // 64

  knn_kernel<<<BB * 8, 256, 0, stream>>>(x, idxbuf);
  zero_kernel<<<1, 128, 0, stream>>>(sums, 128);
  stats_kernel<<<128, 256, 0, stream>>>(x, Wm, idxbuf, sums, sumsq);
  affine_kernel<<<1, 64, 0, stream>>>(sums, sumsq, bnw, bnb, scale, shift);
  out_kernel<<<(BB * NN) / 16 / 8, 256, 0, stream>>>(x, Wm, idxbuf, scale, shift, out);
}